// Net_8289286881434
// MI455X (gfx1250) — compile-verified
//
#include <hip/hip_runtime.h>

// Grouped tiny-linear: y[b,g] = sum_d x[b, g*6+d] * W[g,d] + bias[g]
// B = 4,194,304 rows, 10 groups of 6.  Memory-bound (~0.43 FLOP/byte), so the
// kernel is a single-pass stream; the contraction is mapped onto
// V_WMMA_F32_16X16X4_F32 as Y[16x16] = X[16x60] * Bdense[60x16] (block-diag).

typedef __attribute__((ext_vector_type(2))) float v2f;  // A/B chunk: 2 VGPRs
typedef __attribute__((ext_vector_type(8))) float v8f;  // C/D: 8 VGPRs

#define NG 10               // groups
#define GD 6                // dims per group
#define KTOT (NG * GD)      // 60
#define ROWS_PER_WAVE 16
#define WAVES_PER_BLOCK 8   // 256 threads, wave32

__global__ __launch_bounds__(256)
void grouped_linear_wmma(const float* __restrict__ x,
                         const float* __restrict__ W,     // [10,6] row-major = Wflat[60]
                         const float* __restrict__ bias,  // [10]
                         float* __restrict__ y,           // [B,10]
                         int nRows, int nTiles)
{
    const int lane = threadIdx.x & 31;
    const int wave = threadIdx.x >> 5;
    const int tile = blockIdx.x * WAVES_PER_BLOCK + wave;
    if (tile >= nTiles) return;

    const int rowBase = tile * ROWS_PER_WAVE;
    const int n  = lane & 15;   // column (group id) this lane owns in B/C/D
    const int hi = lane >> 4;   // 0 = low half (K/M 0..), 1 = high half (+2 / +8)

    if (rowBase + ROWS_PER_WAVE <= nRows) {
        // ---- Precompute per-lane B-matrix values for all 15 K-chunks ----
        // Bdense[k][n] = (k/6 == n) ? Wflat[k] : 0   (cols 10..15 are zero)
        // B 4x16 chunk at k0: VGPR j -> lanes 0-15 row k0+j, lanes 16-31 row k0+2+j
        float bm[15][2];
#pragma unroll
        for (int i = 0; i < 15; ++i) {
#pragma unroll
            for (int j = 0; j < 2; ++j) {
                const int k = 4 * i + 2 * hi + j;
                float v = 0.0f;
                if (k < KTOT && (k / GD) == n) v = W[k];
                bm[i][j] = v;
            }
        }

        // ---- A stream + 15 accumulating WMMAs (K = 60) ----
        // A 16x4 f32 layout: lane L<16 holds x[row=L, k0..k0+1] in 2 VGPRs,
        // lanes 16-31 hold k0+2..k0+3.  One aligned b64 load per lane per chunk.
        const float* __restrict__ xrow = x + (size_t)(rowBase + n) * KTOT;
        v8f c = {0.f, 0.f, 0.f, 0.f, 0.f, 0.f, 0.f, 0.f};
#pragma unroll
        for (int i = 0; i < 15; ++i) {
            const int k0 = 4 * i + 2 * hi;
            v2f a = *(const v2f*)(xrow + k0);          // 8B-aligned: 240B row stride, even k
            v2f b;
            b[0] = bm[i][0];
            b[1] = bm[i][1];
            // 8 args: (neg_a, A, neg_b, B, c_mod, C, reuse_a, reuse_b)
            c = __builtin_amdgcn_wmma_f32_16x16x4_f32(
                    false, a, false, b, (short)0, c, false, false);
        }

        // ---- Bias + non-temporal store (write-once stream) ----
        // C/D: VGPR r -> (M = r + 8*hi, N = n); only N < 10 is a real output.
        if (n < NG) {
            const float bn = bias[n];
            const int mBase = hi * 8;
#pragma unroll
            for (int r = 0; r < 8; ++r) {
                const int row = rowBase + mBase + r;
                __builtin_nontemporal_store(c[r] + bn, &y[(size_t)row * NG + n]);
            }
        }
    } else {
        // Tail tile (not hit for B = 4M, but keeps the kernel general):
        // lane -> one row, plain scalar contraction.
        const int row = rowBase + lane;
        if (row < nRows) {
            const float* __restrict__ xr = x + (size_t)row * KTOT;
            for (int g = 0; g < NG; ++g) {
                float acc = bias[g];
                for (int d = 0; d < GD; ++d) acc += xr[g * GD + d] * W[g * GD + d];
                y[(size_t)row * NG + g] = acc;
            }
        }
    }
}

extern "C" void kernel_launch(void* const* d_in, const int* in_sizes, int n_in,
                              void* d_out, int out_size, void* d_ws, size_t ws_size,
                              hipStream_t stream) {
    const float* x    = (const float*)d_in[0];   // [B, 60] fp32
    const float* W    = (const float*)d_in[1];   // [10, 6] fp32
    const float* bias = (const float*)d_in[2];   // [10]   fp32
    float* y          = (float*)d_out;           // [B, 10] fp32

    const int nRows  = in_sizes[0] / KTOT;                        // 4,194,304
    const int nTiles = (nRows + ROWS_PER_WAVE - 1) / ROWS_PER_WAVE;
    const int blocks = (nTiles + WAVES_PER_BLOCK - 1) / WAVES_PER_BLOCK;

    hipLaunchKernelGGL(grouped_linear_wmma, dim3(blocks), dim3(256), 0, stream,
                       x, W, bias, y, nRows, nTiles);
}